// ActorGNN_16784732192966
// MI455X (gfx1250) — compile-verified
//
#include <hip/hip_runtime.h>
#include <hip/hip_bf16.h>

typedef __attribute__((ext_vector_type(2))) float v2f;
typedef __attribute__((ext_vector_type(8))) float v8f;

#define HID 128

// ---------------------------------------------------------------------------
// WMMA f32 GEMM:  C[M, 128] = epilogue( A[M, K] @ B[K, 128] )
// One wave computes one 16x16 output tile via V_WMMA_F32_16X16X4_F32.
// Block = 256 threads = 8 waves -> one block covers a 16-row x 128-col stripe.
//
// Lane layouts (ISA 7.12.2, 32-bit):
//   A 16x4 : lanes 0-15 hold M=lane, {K=0,K=1} in vgpr{0,1}; lanes 16-31 hold {K=2,K=3}
//   B 4x16 : lanes 0-15 hold N=lane, {K=0,K=1};             lanes 16-31 hold {K=2,K=3}
//   C 16x16: vgpr j -> M = j + 8*(lane>=16), N = lane&15
// ---------------------------------------------------------------------------
template <bool RELU, bool ADD_AGG, bool ADD_BIAS>
__global__ __launch_bounds__(256) void gemm_wmma_f32(
    const float* __restrict__ A, const float* __restrict__ B,
    const float* __restrict__ agg, const float* __restrict__ bias,
    float* __restrict__ C, int M, int K) {
  const int lane = threadIdx.x & 31;
  const int wave = threadIdx.x >> 5;       // 0..7 -> N tile
  const int m0   = blockIdx.x * 16;
  const int n0   = wave * 16;
  const int half = lane >> 4;              // 0: K+{0,1}, 1: K+{2,3}
  const int r    = lane & 15;

  int arow = m0 + r;
  if (arow >= M) arow = M - 1;             // clamp (store is guarded); keeps EXEC all-ones
  const float* __restrict__ Arow = A + (size_t)arow * K;

  v8f acc = {};
  for (int k = 0; k < K; k += 4) {
    const int ka = k + 2 * half;
    v2f a, b;
    a.x = Arow[ka];
    a.y = Arow[ka + 1];
    b.x = B[(size_t)ka * HID + n0 + r];
    b.y = B[(size_t)(ka + 1) * HID + n0 + r];
    // D = A*B + C, fp32 end to end
    acc = __builtin_amdgcn_wmma_f32_16x16x4_f32(
        /*neg_a=*/false, a, /*neg_b=*/false, b,
        /*c_mod=*/(short)0, acc, /*reuse_a=*/false, /*reuse_b=*/false);
  }

  const int col = n0 + r;
  float bval = 0.0f;
  if (ADD_BIAS) bval = bias[col];
#pragma unroll
  for (int j = 0; j < 8; ++j) {
    const int row = m0 + j + 8 * half;
    if (row < M) {
      float v = acc[j];
      if (ADD_AGG) v += agg[(size_t)row * HID + col];
      if (ADD_BIAS) v += bval;
      if (RELU) v = fmaxf(v, 0.0f);
      C[(size_t)row * HID + col] = v;
    }
  }
}

// ---------------------------------------------------------------------------
// agg = 0
// ---------------------------------------------------------------------------
__global__ __launch_bounds__(256) void zero_f32(float4* __restrict__ p, int n4) {
  const int i = blockIdx.x * blockDim.x + threadIdx.x;
  if (i < n4) p[i] = float4{0.0f, 0.0f, 0.0f, 0.0f};
}

// ---------------------------------------------------------------------------
// Edge scatter-add: agg[dst[e]] += hw[src[e]]   (128 floats per edge)
// One wave per edge; each lane moves one float4 and issues 4 fp32 hardware
// atomics via inline CDNA5 asm: guaranteed global_atomic_add_f32 (no CAS
// loop), device scope so the RMW executes at the L2 atomic units (agg lines
// are shared across workgroups). Fire-and-forget: tracked by STOREcnt,
// drained by s_endpgm's implicit wait-idle.
// ---------------------------------------------------------------------------
__global__ __launch_bounds__(256) void edge_scatter(
    const float* __restrict__ hw, const long long* __restrict__ src,
    const long long* __restrict__ dst, float* __restrict__ agg, int E) {
  const int e    = (int)((blockIdx.x * (size_t)blockDim.x + threadIdx.x) >> 5);
  const int lane = threadIdx.x & 31;
  if (e >= E) return;
  const long long s = src[e];
  const long long d = dst[e];
  const float4 v = *((const float4*)(hw + (size_t)s * HID) + lane);
  float* o = agg + (size_t)d * HID + lane * 4;
  asm volatile(
      "global_atomic_add_f32 %0, %1, off scope:SCOPE_DEV\n\t"
      "global_atomic_add_f32 %0, %2, off offset:4 scope:SCOPE_DEV\n\t"
      "global_atomic_add_f32 %0, %3, off offset:8 scope:SCOPE_DEV\n\t"
      "global_atomic_add_f32 %0, %4, off offset:12 scope:SCOPE_DEV"
      :
      : "v"(o), "v"(v.x), "v"(v.y), "v"(v.z), "v"(v.w)
      : "memory");
}

// ---------------------------------------------------------------------------
// logits[i] = dot(h[i,:], W_out[:,0]) + b_out   — one wave per node
// ---------------------------------------------------------------------------
__global__ __launch_bounds__(256) void out_dot(
    const float* __restrict__ h, const float* __restrict__ Wout,
    const float* __restrict__ bout, float* __restrict__ logits, int M) {
  const int node = (int)((blockIdx.x * (size_t)blockDim.x + threadIdx.x) >> 5);
  const int lane = threadIdx.x & 31;
  if (node >= M) return;
  const float* __restrict__ row = h + (size_t)node * HID;
  float s = 0.0f;
#pragma unroll
  for (int j = 0; j < HID / 32; ++j) s += row[lane + 32 * j] * Wout[lane + 32 * j];
#pragma unroll
  for (int off = 16; off > 0; off >>= 1) s += __shfl_down(s, off, 32);
  if (lane == 0) logits[node] = s + bout[0];
}

// ---------------------------------------------------------------------------
// Host-side orchestration (graph-capture safe: kernels on `stream` only)
// ---------------------------------------------------------------------------
extern "C" void kernel_launch(void* const* d_in, const int* in_sizes, int n_in,
                              void* d_out, int out_size, void* d_ws, size_t ws_size,
                              hipStream_t stream) {
  const float*      x      = (const float*)d_in[0];       // [M, 16]
  const long long*  edge   = (const long long*)d_in[1];   // [2, E] int64
  const float*      W_in   = (const float*)d_in[2];       // [16, 128]
  const float*      W_self = (const float*)d_in[3];       // [3, 128, 128]
  const float*      W_nbr  = (const float*)d_in[4];       // [3, 128, 128]
  const float*      bvec   = (const float*)d_in[5];       // [3, 128]
  const float*      W_out  = (const float*)d_in[6];       // [128, 1]
  const float*      b_out  = (const float*)d_in[7];       // [1]
  float*            out    = (float*)d_out;               // [M]

  const int F_IN = 16;
  const int M = in_sizes[0] / F_IN;                       // 10000
  const int E = in_sizes[1] / 2;                          // 640000
  const int L = in_sizes[3] / (HID * HID);                // 3

  // Workspace: h0, h1, hw, agg — each M*128 floats (~5.1 MB), L2-resident.
  float* h0  = (float*)d_ws;
  float* h1  = h0 + (size_t)M * HID;
  float* hw  = h1 + (size_t)M * HID;
  float* agg = hw + (size_t)M * HID;

  const int mtiles = (M + 15) / 16;                       // 625 blocks, 8 waves each
  const dim3 blk(256);

  // h0 = relu(x @ W_in)
  gemm_wmma_f32<true, false, false><<<mtiles, blk, 0, stream>>>(
      x, W_in, nullptr, nullptr, h0, M, F_IN);

  float* hc = h0;
  float* hn = h1;
  const int n4 = M * HID / 4;
  const int zblocks = (n4 + 255) / 256;
  const int sblocks = (int)(((size_t)E * 32 + 255) / 256);

  for (int l = 0; l < L; ++l) {
    // hw = hc @ W_nbr[l]   (gather commutes with the linear map)
    gemm_wmma_f32<false, false, false><<<mtiles, blk, 0, stream>>>(
        hc, W_nbr + (size_t)l * HID * HID, nullptr, nullptr, hw, M, HID);
    // agg = 0
    zero_f32<<<zblocks, blk, 0, stream>>>((float4*)agg, n4);
    // agg[dst] += hw[src]
    edge_scatter<<<sblocks, blk, 0, stream>>>(hw, edge, edge + E, agg, E);
    // hn = relu(hc @ W_self[l] + agg + b[l])
    gemm_wmma_f32<true, true, true><<<mtiles, blk, 0, stream>>>(
        hc, W_self + (size_t)l * HID * HID, agg, bvec + (size_t)l * HID, hn, M, HID);
    float* t = hc; hc = hn; hn = t;
  }

  // logits = h @ W_out + b_out
  const int oblocks = (int)(((size_t)M * 32 + 255) / 256);
  out_dot<<<oblocks, blk, 0, stream>>>(hc, W_out, b_out, out, M);
}